// PrivacyAwareTokenPruning_55301998903580
// MI455X (gfx1250) — compile-verified
//
#include <hip/hip_runtime.h>
#include <hip/hip_bf16.h>

// ---------------------------------------------------------------------------
// PrivacyAwareTokenPruning for MI455X (gfx1250, wave32)
//   out[b, 0..k-1, :] = seq[b, topk_idx[b, i], :]   (descending-weight order)
//   out[b, k,      :] = 0.05 * mean(pruned tokens)
// Memory-bound: ~256MB read + ~128MB write -> ~16.5us floor @ 23.3 TB/s.
// ---------------------------------------------------------------------------

typedef __attribute__((ext_vector_type(2)))  float    v2f;
typedef __attribute__((ext_vector_type(4)))  float    v4f;
typedef __attribute__((ext_vector_type(8)))  float    v8f;

#define SEGS 4   // token-range segments per (batch, d-tile) -> 4x wave count

// ---------------------------------------------------------------------------
// Kernel 1: exact top-k ordering via in-LDS bitonic sort of (value,index) keys.
// One block per batch row; N power of two (4096). XOR-pair formulation keeps
// the index math to bit ops (no integer division).
// Key = (monotonic_u32(value) << 32) | (0xFFFFFFFF - index); descending sort
// gives larger value first, ties -> smaller index first (jax.lax.top_k order).
// ---------------------------------------------------------------------------
__global__ void topk_sort_kernel(const float* __restrict__ attn,
                                 unsigned int* __restrict__ order, int N) {
    extern __shared__ unsigned long long keys[];
    const int b = blockIdx.x;
    const int tid = threadIdx.x;
    const int nthr = blockDim.x;
    const float* w = attn + (size_t)b * N;

    for (int i = tid; i < N; i += nthr) {
        unsigned int bits = __float_as_uint(w[i]);
        unsigned int mono = (bits & 0x80000000u) ? ~bits : (bits | 0x80000000u);
        keys[i] = ((unsigned long long)mono << 32) |
                  (unsigned long long)(0xFFFFFFFFu - (unsigned int)i);
    }

    for (int size = 2; size <= N; size <<= 1) {
        for (int stride = size >> 1; stride > 0; stride >>= 1) {
            __syncthreads();
            for (int i = tid; i < N; i += nthr) {
                const int j = i ^ stride;
                if (j > i) {
                    const bool desc = ((i & size) == 0);  // descending segment
                    unsigned long long a = keys[i];
                    unsigned long long c = keys[j];
                    if ((a < c) == desc) { keys[i] = c; keys[j] = a; }
                }
            }
        }
    }
    __syncthreads();

    for (int i = tid; i < N; i += nthr)
        order[(size_t)b * N + i] =
            0xFFFFFFFFu - (unsigned int)(keys[i] & 0xFFFFFFFFull);
}

// ---------------------------------------------------------------------------
// Kernel 2: gather selected rows. One block per (token slot, batch);
// 256 threads x float4 = one 4KB row, b128-coalesced. Data is read-once /
// write-once -> non-temporal hints keep it out of the way of the pruned half.
// ---------------------------------------------------------------------------
__global__ void gather_kernel(const float* __restrict__ seq,
                              const unsigned int* __restrict__ order,
                              float* __restrict__ out, int N, int D, int k) {
    const int i = blockIdx.x;   // selected slot 0..k-1
    const int b = blockIdx.y;
    const unsigned int tok = order[(size_t)b * N + i];
    const v4f* __restrict__ src =
        (const v4f*)(seq + ((size_t)b * N + tok) * (size_t)D);
    v4f* __restrict__ dst =
        (v4f*)(out + ((size_t)b * (k + 1) + i) * (size_t)D);
    const int nvec = D >> 2;
    for (int v = threadIdx.x; v < nvec; v += blockDim.x) {
        v4f val = __builtin_nontemporal_load(&src[v]);
        __builtin_nontemporal_store(val, &dst[v]);
    }
}

// ---------------------------------------------------------------------------
// Kernel 3: pruned-token sums via WMMA. One wave per (segment, batch, 16-col
// d-tile). A = ones(16x4), so D = A x B + C accumulates 4-token column sums;
// invariant to the B-operand K<->lane-half permutation. Row M=0 of C (lanes
// 0..15, C VGPR0) carries the sums. Branchless main loop: 8 prefetched index
// regs -> 8 value loads (u32 offsets, GVS) -> 4 WMMAs on independent
// accumulators (hides the WMMA D->C RAW chain). Guarded tail is dead when
// pcount % (SEGS*16) == 0 (true here: 2048 % 64 == 0).
// ---------------------------------------------------------------------------
__global__ void __launch_bounds__(32)
rem_wmma_kernel(const float* __restrict__ seq,
                const unsigned int* __restrict__ order,
                float* __restrict__ partials,
                int N, int D, int k, int B) {
    const int tilesPerBatch = D >> 4;
    const int wavesPerSeg = B * tilesPerBatch;
    const int seg = blockIdx.x / wavesPerSeg;
    const int rem = blockIdx.x % wavesPerSeg;
    const int b  = rem / tilesPerBatch;
    const int d0 = (rem % tilesPerBatch) << 4;
    const int lane = threadIdx.x;

    const unsigned int* __restrict__ pruned = order + (size_t)b * N + k;
    const int pcount = N - k;
    const int chunk  = (pcount + SEGS - 1) / SEGS;
    const int segBeg = seg * chunk;
    int segEnd = segBeg + chunk; if (segEnd > pcount) segEnd = pcount;
    const int segLen = (segEnd > segBeg) ? (segEnd - segBeg) : 0;
    const int mainEnd = segBeg + (segLen & ~15);

    const float* __restrict__ sb = seq + (size_t)b * N * (size_t)D;
    const unsigned du   = (unsigned)D;
    const unsigned colu = (unsigned)(d0 + (lane & 15));
    const int koff = (lane >> 4) * 2;      // lanes 0-15: K=0,1 ; 16-31: K=2,3

    const v2f a_ones = {1.0f, 1.0f};
    v8f acc[4] = {v8f{}, v8f{}, v8f{}, v8f{}};

    unsigned tnext[8];
    if (segBeg < mainEnd) {
#pragma unroll
        for (int g = 0; g < 4; ++g) {
            tnext[2 * g]     = pruned[segBeg + g * 4 + koff];
            tnext[2 * g + 1] = pruned[segBeg + g * 4 + koff + 1];
        }
    }
    for (int j = segBeg; j < mainEnd; j += 16) {
        unsigned tcur[8];
#pragma unroll
        for (int q = 0; q < 8; ++q) tcur[q] = tnext[q];
        if (j + 16 < mainEnd) {   // prefetch next iteration's indices
#pragma unroll
            for (int g = 0; g < 4; ++g) {
                tnext[2 * g]     = pruned[j + 16 + g * 4 + koff];
                tnext[2 * g + 1] = pruned[j + 16 + g * 4 + koff + 1];
            }
        }
        float x[8];
#pragma unroll
        for (int q = 0; q < 8; ++q) x[q] = sb[tcur[q] * du + colu];
#pragma unroll
        for (int g = 0; g < 4; ++g) {
            v2f bm; bm.x = x[2 * g]; bm.y = x[2 * g + 1];
            acc[g] = __builtin_amdgcn_wmma_f32_16x16x4_f32(
                false, a_ones, false, bm, (short)0, acc[g], false, false);
        }
    }

    // guarded tail (never taken for N=4096, k=2048)
    for (int j = mainEnd; j < segEnd; j += 4) {
        const int s0 = j + koff, s1 = s0 + 1;
        v2f bm;
        bm.x = (s0 < segEnd) ? sb[pruned[s0] * du + colu] : 0.0f;
        bm.y = (s1 < segEnd) ? sb[pruned[s1] * du + colu] : 0.0f;
        acc[0] = __builtin_amdgcn_wmma_f32_16x16x4_f32(
            false, a_ones, false, bm, (short)0, acc[0], false, false);
    }

    v8f asum = (acc[0] + acc[1]) + (acc[2] + acc[3]);
    if (lane < 16) {
        // C tile row M=0 lives in C VGPR0 on lanes 0..15 (N = lane).
        partials[((size_t)seg * B + b) * (size_t)D + d0 + lane] = asum[0];
    }
}

// ---------------------------------------------------------------------------
// Kernel 4: deterministic segment reduction -> final mixup token row.
// ---------------------------------------------------------------------------
__global__ void finalize_kernel(const float* __restrict__ partials,
                                float* __restrict__ out,
                                int B, int D, int k, float scale) {
    const int idx = blockIdx.x * blockDim.x + threadIdx.x;   // over B*D
    if (idx >= B * D) return;
    const int b = idx / D;
    const int d = idx % D;
    float s = 0.0f;
#pragma unroll
    for (int q = 0; q < SEGS; ++q)
        s += partials[((size_t)q * B + b) * (size_t)D + d];
    out[((size_t)b * (k + 1) + k) * (size_t)D + d] = s * scale;
}

// ---------------------------------------------------------------------------
// Launch
// ---------------------------------------------------------------------------
extern "C" void kernel_launch(void* const* d_in, const int* in_sizes, int n_in,
                              void* d_out, int out_size, void* d_ws, size_t ws_size,
                              hipStream_t stream) {
    const float* seq  = (const float*)d_in[0];
    const float* attn = (const float*)d_in[1];
    float* out = (float*)d_out;

    const long long sseq  = in_sizes[0];         // B*N*D
    const long long sattn = in_sizes[1];         // B*N
    const int D = (int)(sseq / sattn);           // 1024
    const int B = (int)((long long)out_size / D - sattn / 2);  // 16
    const int N = (int)(sattn / B);              // 4096
    int k = N / 2;                               // PRUNE_RATIO = 0.5
    if (k < 1) k = 1;

    unsigned int* order = (unsigned int*)d_ws;                      // B*N u32
    float* partials = (float*)((char*)d_ws +
                               (size_t)B * N * sizeof(unsigned));   // SEGS*B*D f32

    // 1) exact descending ranking per batch (LDS bitonic sort, 32KB/block)
    topk_sort_kernel<<<B, 1024, (size_t)N * sizeof(unsigned long long), stream>>>(
        attn, order, N);

    // 2) gather the selected half (128 MB, b128-coalesced, non-temporal)
    dim3 ggrid(k, B);
    gather_kernel<<<ggrid, 256, 0, stream>>>(seq, order, out, N, D, k);

    // 3) pruned-token sums via WMMA (other 128 MB), 4 segments for occupancy
    rem_wmma_kernel<<<SEGS * B * (D / 16), 32, 0, stream>>>(
        seq, order, partials, N, D, k, B);

    // 4) deterministic reduction of segment partials -> mixup token
    const float scale = 0.05f / ((float)(N - k) + 1e-10f);
    const int total = B * D;
    finalize_kernel<<<(total + 255) / 256, 256, 0, stream>>>(
        partials, out, B, D, k, scale);
}